// GCNOGBLayer_9586367005318
// MI455X (gfx1250) — compile-verified
//
#include <hip/hip_runtime.h>

// fp32 WMMA fragments (wave32): A/B = 2 VGPRs, C/D = 8 VGPRs
typedef __attribute__((ext_vector_type(2))) float v2f;
typedef __attribute__((ext_vector_type(8))) float v8f;

#define N_NODES 50000
#define N_EDGES 800000
#define DIM 64
#define WAVES_PER_BLOCK 8

__device__ __forceinline__ void atomic_add_f32(float* p, float v) {
    // non-returning agent-scope fp32 atomic -> global_atomic_add_f32
    __hip_atomic_fetch_add(p, v, __ATOMIC_RELAXED, __HIP_MEMORY_SCOPE_AGENT);
}

// Pair-contiguous LDS weight layout:
//   WtP[(k>>1)*128 + n*2 + (k&1)] = W[n*64 + k]   (B[k][n] = W[n][k])
// B fragment for WMMA K-group kk = 4k + 2*half is then ONE aligned ds_load_b64:
//   b = float2 @ WtP[(2k+half)*128 + 32n + 2*lcol]
__device__ __forceinline__ void stage_weights(const float* __restrict__ W,
                                              float* __restrict__ WtP) {
    for (int idx = threadIdx.x; idx < DIM * DIM; idx += blockDim.x) {
        int n = idx >> 6, k = idx & 63;
        WtP[(k >> 1) * 128 + n * 2 + (k & 1)] = W[idx];
    }
    __syncthreads();
}

// ---------------------------------------------------------------------------
// Kernel 1: h = node_feats @ Wn^T + bn (stored to ws), out = relu(h+res_w)/degs
// One wave computes one 16x64 output tile via V_WMMA_F32_16X16X4_F32.
// ---------------------------------------------------------------------------
__global__ void __launch_bounds__(256) gcn_node_kernel(
    const float* __restrict__ node_feats, const float* __restrict__ degs,
    const float* __restrict__ Wn, const float* __restrict__ bn,
    const float* __restrict__ res_w, float* __restrict__ h,
    float* __restrict__ out, int n_tiles)
{
    __shared__ float WtP[DIM * DIM];
    stage_weights(Wn, WtP);

    const int lane = threadIdx.x & 31;
    const int wave = threadIdx.x >> 5;
    const int half = lane >> 4;   // 0 or 1 (lane group)
    const int lcol = lane & 15;

    const int tile = blockIdx.x * WAVES_PER_BLOCK + wave;
    if (tile >= n_tiles) return;  // wave-uniform exit: EXEC stays all-1s
    const int R = tile * 16;

    v8f acc[4] = {v8f{}, v8f{}, v8f{}, v8f{}};

    // A fragment: lane holds row (R+lcol), elements K = 4k + 2*half + {0,1}
    const float* arow = node_feats + (size_t)(R + lcol) * DIM + 2 * half;
    const float* bbase = WtP + half * 128 + 2 * lcol;

    #pragma unroll
    for (int k = 0; k < 16; ++k) {
        v2f a = *(const v2f*)(arow + 4 * k);
        const float* bk = bbase + k * 256;
        #pragma unroll
        for (int n = 0; n < 4; ++n) {
            v2f b = *(const v2f*)(bk + 32 * n);   // single aligned ds_load_b64
            acc[n] = __builtin_amdgcn_wmma_f32_16x16x4_f32(
                false, a, false, b, (short)0, acc[n], false, false);
        }
    }

    // Epilogue: C layout -> lane, VGPR i holds (M = i + 8*half, N = lcol)
    #pragma unroll
    for (int i = 0; i < 8; ++i) {
        const int row = R + i + 8 * half;
        const float dg = degs[row];
        #pragma unroll
        for (int n = 0; n < 4; ++n) {
            const int col = 16 * n + lcol;
            const float hv = acc[n][i] + bn[col];
            h[(size_t)row * DIM + col] = hv;
            out[(size_t)row * DIM + col] = fmaxf(hv + res_w[col], 0.0f) / dg;
        }
    }
}

// ---------------------------------------------------------------------------
// Kernel 2: ep = edge_feats @ We^T + be; m = norm*relu(h[src]+ep);
//           out[dst] += m  (atomic scatter-add / segment_sum)
// One wave computes TWO 16x64 tiles (32 edges): each B fragment feeds 2 WMMAs.
// 800000/32 = 25000 wave-pairs exactly -> no tail, EXEC all-1s everywhere.
// ---------------------------------------------------------------------------
__global__ void __launch_bounds__(256) gcn_edge_kernel(
    const float* __restrict__ edge_feats, const float* __restrict__ norm,
    const int* __restrict__ src, const int* __restrict__ dst,
    const float* __restrict__ We, const float* __restrict__ be,
    const float* __restrict__ h, float* __restrict__ out, int n_pairs)
{
    __shared__ float WtP[DIM * DIM];
    stage_weights(We, WtP);

    const int lane = threadIdx.x & 31;
    const int wave = threadIdx.x >> 5;
    const int half = lane >> 4;
    const int lcol = lane & 15;

    const int pair = blockIdx.x * WAVES_PER_BLOCK + wave;
    if (pair >= n_pairs) return;
    const int R = pair * 32;       // 32 edges per wave

    v8f acc0[4] = {v8f{}, v8f{}, v8f{}, v8f{}};
    v8f acc1[4] = {v8f{}, v8f{}, v8f{}, v8f{}};

    const float* arow0 = edge_feats + (size_t)(R + lcol) * DIM + 2 * half;
    const float* arow1 = arow0 + 16 * DIM;
    const float* bbase = WtP + half * 128 + 2 * lcol;

    #pragma unroll
    for (int k = 0; k < 16; ++k) {
        v2f a0 = *(const v2f*)(arow0 + 4 * k);
        v2f a1 = *(const v2f*)(arow1 + 4 * k);
        const float* bk = bbase + k * 256;
        #pragma unroll
        for (int n = 0; n < 4; ++n) {
            v2f b = *(const v2f*)(bk + 32 * n);   // one ds_load_b64, two WMMAs
            acc0[n] = __builtin_amdgcn_wmma_f32_16x16x4_f32(
                false, a0, false, b, (short)0, acc0[n], false, false);
            acc1[n] = __builtin_amdgcn_wmma_f32_16x16x4_f32(
                false, a1, false, b, (short)0, acc1[n], false, false);
        }
    }

    // Epilogue per tile: gather h[src], +be, relu, *norm, scatter-add out[dst]
    #pragma unroll
    for (int t = 0; t < 2; ++t) {
        const v8f* acc = t ? acc1 : acc0;
        const int Rt = R + t * 16;
        #pragma unroll
        for (int i = 0; i < 8; ++i) {
            const int e = Rt + i + 8 * half;
            const int s = src[e];
            const int d = dst[e];
            const float nr = norm[e];
            #pragma unroll
            for (int n = 0; n < 4; ++n) {
                const int col = 16 * n + lcol;
                const float ep = acc[n][i] + be[col];
                const float hv = h[(size_t)s * DIM + col];
                const float m = fmaxf(hv + ep, 0.0f) * nr;
                atomic_add_f32(out + (size_t)d * DIM + col, m);
            }
        }
    }
}

extern "C" void kernel_launch(void* const* d_in, const int* in_sizes, int n_in,
                              void* d_out, int out_size, void* d_ws, size_t ws_size,
                              hipStream_t stream) {
    const float* node_feats = (const float*)d_in[0];
    const float* edge_feats = (const float*)d_in[1];
    const float* degs       = (const float*)d_in[2];
    const float* norm       = (const float*)d_in[3];
    const int*   src        = (const int*)d_in[4];
    const int*   dst        = (const int*)d_in[5];
    const float* Wn         = (const float*)d_in[6];
    const float* bn         = (const float*)d_in[7];
    const float* We         = (const float*)d_in[8];
    const float* be         = (const float*)d_in[9];
    const float* res_w      = (const float*)d_in[10];

    float* out = (float*)d_out;
    float* h   = (float*)d_ws;  // N*D*4 = 12.8 MB scratch for transformed node feats

    const int node_tiles = N_NODES / 16;  // 3125 (exact)
    const int edge_pairs = N_EDGES / 32;  // 25000 (exact)

    // Kernel 1 writes out = res term; kernel 2 accumulates agg on top (stream-ordered).
    gcn_node_kernel<<<(node_tiles + WAVES_PER_BLOCK - 1) / WAVES_PER_BLOCK, 256, 0, stream>>>(
        node_feats, degs, Wn, bn, res_w, h, out, node_tiles);
    gcn_edge_kernel<<<(edge_pairs + WAVES_PER_BLOCK - 1) / WAVES_PER_BLOCK, 256, 0, stream>>>(
        edge_feats, norm, src, dst, We, be, h, out, edge_pairs);
}